// PairwiseScores_Cls_154618822964
// MI455X (gfx1250) — compile-verified
//
#include <hip/hip_runtime.h>

#define DIM 768
#define HID 128
#define BQ  256
#define BP  1024

typedef __attribute__((ext_vector_type(2))) float v2f;
typedef __attribute__((ext_vector_type(8))) float v8f;

// ---------------------------------------------------------------------------
// Kernel 1: H(M x 128) = X(M x 768) @ W1[:, koff:koff+768]^T  (+ optional bias)
// One wave32 per 16x16 tile of H, accumulated with V_WMMA_F32_16X16X4_F32.
// Fragment layouts (CDNA5 ISA 7.12.2, f32 16x16x4):
//   A 16x4 : lane&15 = M row; lanes<16 hold K=0,1 ; lanes>=16 hold K=2,3
//   B 4x16 : lane&15 = N col; lanes<16 hold K=0,1 ; lanes>=16 hold K=2,3
//   C 16x16: VGPR r -> M=r (lanes 0-15) / M=r+8 (lanes 16-31), N = lane&15
// B[k][n] = W1[n][koff+k], so both fragments are contiguous float2 loads.
// ---------------------------------------------------------------------------
__global__ __launch_bounds__(128) void proj_wmma(const float* __restrict__ X,
                                                 const float* __restrict__ W1,
                                                 const float* __restrict__ bias,
                                                 float* __restrict__ H,
                                                 int M, int koff) {
  const int lane = threadIdx.x & 31;
  const int wave = blockIdx.x * (blockDim.x >> 5) + (threadIdx.x >> 5);
  const int tiles_n = HID / 16;                 // 8 tiles along N=128
  const int tm = (wave / tiles_n) * 16;
  const int tn = (wave % tiles_n) * 16;
  if (tm >= M) return;                          // wave-uniform guard

  const int l15 = lane & 15;
  const int kb  = (lane >> 4) << 1;             // 0 (lanes 0-15) or 2 (16-31)
  const float* arow = X  + (size_t)(tm + l15) * DIM;
  const float* brow = W1 + (size_t)(tn + l15) * (2 * DIM) + koff;

  v8f acc = {};
#pragma unroll 4
  for (int k = 0; k < DIM; k += 4) {
    v2f a = *(const v2f*)(arow + k + kb);       // A[m][k+kb], A[m][k+kb+1]
    v2f b = *(const v2f*)(brow + k + kb);       // W1[n][koff+k+kb], +1
    acc = __builtin_amdgcn_wmma_f32_16x16x4_f32(
        /*neg_a=*/false, a, /*neg_b=*/false, b,
        /*c_mod=*/(short)0, acc, /*reuse_a=*/false, /*reuse_b=*/false);
  }

  const float bv   = bias ? bias[tn + l15] : 0.0f;   // bias is per-N
  const int mbase  = tm + ((lane >> 4) << 3);        // +8 for upper half-wave
  const int nout   = tn + l15;
#pragma unroll
  for (int r = 0; r < 8; ++r)
    H[(size_t)(mbase + r) * HID + nout] = acc[r] + bv;
}

// ---------------------------------------------------------------------------
// Kernel 2: out[q][p] = b2 + sum_h relu(qh[q][h] + dhb[p][h]) * w2[h]
// (dhb already contains +b1). 16 q x 64 p tile per 256-thread block.
// LDS rows padded to 129 floats so the 16 lanes reading ds[tp + 16j][h]
// (stride 129 % 64 banks = 1) are conflict-free; qs reads are broadcasts.
// ---------------------------------------------------------------------------
#define TQ 16
#define TP 64

__global__ __launch_bounds__(256) void pair_score(const float* __restrict__ qh,
                                                  const float* __restrict__ dhb,
                                                  const float* __restrict__ W2,
                                                  const float* __restrict__ b2,
                                                  float* __restrict__ out) {
  __shared__ float qs[TQ][HID + 1];
  __shared__ float ds[TP][HID + 1];
  __shared__ float ws[HID];

  const int tid = threadIdx.x;
  const int qb  = blockIdx.y * TQ;
  const int pb  = blockIdx.x * TP;

  for (int i = tid; i < TQ * HID; i += 256) {
    int r = i >> 7, c = i & (HID - 1);
    qs[r][c] = qh[(size_t)(qb + r) * HID + c];
  }
  for (int i = tid; i < TP * HID; i += 256) {
    int r = i >> 7, c = i & (HID - 1);
    ds[r][c] = dhb[(size_t)(pb + r) * HID + c];
  }
  if (tid < HID) ws[tid] = W2[tid];
  __syncthreads();

  const int tq = tid >> 4;       // 0..15  -> q row
  const int tp = tid & 15;       // 0..15  -> p cols tp, tp+16, tp+32, tp+48
  const float* qr = qs[tq];
  const float* d0 = ds[tp];
  const float* d1 = ds[tp + 16];
  const float* d2 = ds[tp + 32];
  const float* d3 = ds[tp + 48];

  float a0 = 0.f, a1 = 0.f, a2 = 0.f, a3 = 0.f;
#pragma unroll 4
  for (int h = 0; h < HID; ++h) {
    float a = qr[h];             // LDS broadcast across 16 lanes
    float w = ws[h];
    a0 = fmaf(w, fmaxf(a + d0[h], 0.f), a0);
    a1 = fmaf(w, fmaxf(a + d1[h], 0.f), a1);
    a2 = fmaf(w, fmaxf(a + d2[h], 0.f), a2);
    a3 = fmaf(w, fmaxf(a + d3[h], 0.f), a3);
  }

  const float bb = b2[0];
  const size_t ro = (size_t)(qb + tq) * BP + pb + tp;
  out[ro]      = a0 + bb;        // lanes 0-15 write 16 consecutive floats
  out[ro + 16] = a1 + bb;
  out[ro + 32] = a2 + bb;
  out[ro + 48] = a3 + bb;
}

extern "C" void kernel_launch(void* const* d_in, const int* in_sizes, int n_in,
                              void* d_out, int out_size, void* d_ws, size_t ws_size,
                              hipStream_t stream) {
  const float* query = (const float*)d_in[0];   // (256, 768)
  const float* doc   = (const float*)d_in[1];   // (1024, 768)
  const float* W1    = (const float*)d_in[2];   // (128, 1536)
  const float* b1    = (const float*)d_in[3];   // (128,)
  const float* W2    = (const float*)d_in[4];   // (1, 128)
  const float* b2    = (const float*)d_in[5];   // (1,)
  float* out = (float*)d_out;                   // (256, 1024)

  float* qh = (float*)d_ws;                     // 256*128 floats
  float* dh = qh + (size_t)BQ * HID;            // 1024*128 floats (total 640 KB)

  // qh = query @ W1[:, :768]^T          (16 m-tiles * 8 n-tiles = 128 waves)
  proj_wmma<<<dim3(32), 128, 0, stream>>>(query, W1, nullptr, qh, BQ, 0);
  // dh = doc @ W1[:, 768:]^T + b1       (64 m-tiles * 8 n-tiles = 512 waves)
  proj_wmma<<<dim3(128), 128, 0, stream>>>(doc, W1, b1, dh, BP, DIM);
  // out = relu-pairwise dot with w2, + b2
  pair_score<<<dim3(BP / TP, BQ / TQ), 256, 0, stream>>>(qh, dh, W2, b2, out);
}